// Discriminator_18038862643864
// MI455X (gfx1250) — compile-verified
//
#include <hip/hip_runtime.h>

typedef __attribute__((ext_vector_type(2))) float v2f;
typedef __attribute__((ext_vector_type(8))) float v8f;

// ---------------------------------------------------------------------------
// Kernel A: conv1 (4D, k=3, s=2, p=1) + bias + relu as implicit GEMM on the
// fp32 WMMA pipe (V_WMMA_F32_16X16X4_F32, exact fp32 math).
//   x : [64,4,8,8,8,16]  -> h1 : [64,2,4,4,4,8]
//   M = 64*512 = 32768 output positions (16 per wave)
//   K-dim of each WMMA = the 4 input channels (no padding, no div/mod)
//   steps = 81 spatial taps (3^4), strength-reduced nested loops
//   N = 16 (Cout=2 valid, rest read a zero-padded LDS region)
// Gathers use a uniform SGPR base + unsigned 32-bit VGPR offset so the
// backend can use GVS addressing (no per-lane 64-bit address formation).
// ---------------------------------------------------------------------------
__global__ __launch_bounds__(128) void conv1_wmma_f32(
    const float* __restrict__ x,   // [64,4,8,8,8,16]
    const float* __restrict__ w1,  // [2,4,3,3,3,3] == [n][cin*81 + tap]
    const float* __restrict__ b1,  // [2]
    float* __restrict__ h1)        // [64,2,4,4,4,8]
{
    // [0..648) = real weights, [648..816) = zeros (pad rows for N >= 2)
    __shared__ float wlds[816];
    const int tid = threadIdx.x;
    for (int i = tid; i < 816; i += 128)
        wlds[i] = (i < 648) ? w1[i] : 0.0f;
    __syncthreads();

    const int wid  = tid >> 5;
    const int lane = tid & 31;
    const int tile = blockIdx.x * 4 + wid;       // 2048 M-tiles of 16
    const int row  = lane & 15;                  // A-fragment row (M)
    const int cin0 = (lane >> 4) << 1;           // K slots: cin0, cin0+1
    const int n    = lane & 15;                  // B/C column (N)

    // Decompose this lane's output position m (for the A gather)
    const int m   = tile * 16 + row;
    const int b   = m >> 9;                      // / 512
    const int rem = m & 511;
    const int i1b = ((rem >> 7)     ) * 2 - 1;   // od1*2 - 1
    const int i2b = ((rem >> 5) & 3 ) * 2 - 1;
    const int i3b = ((rem >> 3) & 3 ) * 2 - 1;
    const int i4b = ( rem       & 7 ) * 2 - 1;

    // per-lane element base into x (batch + first channel of this lane's pair)
    const unsigned lbase = (unsigned)(b * 32768 + cin0 * 8192);

    // B-fragment LDS base: real rows for n<2, zero tail for n>=2
    const int wbase = ((n < 2) ? n * 324 : 648) + cin0 * 81;

    v8f acc = {};
    int tap = 0;
    for (int k1 = 0; k1 < 3; ++k1) {
        const int  i1  = i1b + k1;
        const bool ok1 = (unsigned)i1 < 8u;
        const int  s1  = i1 * 1024;
        for (int k2 = 0; k2 < 3; ++k2) {
            const int  i2  = i2b + k2;
            const bool ok2 = ok1 & ((unsigned)i2 < 8u);
            const int  s2  = s1 + i2 * 128;
            for (int k3 = 0; k3 < 3; ++k3) {
                const int  i3  = i3b + k3;
                const bool ok3 = ok2 & ((unsigned)i3 < 8u);
                const unsigned sbase = lbase + (unsigned)(s2 + i3 * 16);
#pragma unroll
                for (int k4 = 0; k4 < 3; ++k4) {
                    const int  i4 = i4b + k4;
                    const bool ok = ok3 & ((unsigned)i4 < 16u);
                    // branch-free OOB: clamp offset into this lane's base
                    const unsigned uoff = ok ? (sbase + (unsigned)i4) : lbase;
                    const float va = x[uoff];             // K = cin0
                    const float vb = x[uoff + 8192u];     // K = cin0+1
                    v2f a, bb;
                    a[0]  = ok ? va : 0.0f;
                    a[1]  = ok ? vb : 0.0f;
                    bb[0] = wlds[wbase + tap];        // B[cin0][n]
                    bb[1] = wlds[wbase + tap + 81];   // B[cin0+1][n]
                    acc = __builtin_amdgcn_wmma_f32_16x16x4_f32(
                        false, a, false, bb, (short)0, acc, false, false);
                    ++tap;
                }
            }
        }
    }

    // C layout: lane n = N; VGPR r holds M = r (lanes 0-15) / r+8 (16-31)
    if (n < 2) {
        const float bias  = b1[n];
        const int   mhalf = (lane >> 4) << 3;    // 0 or 8
#pragma unroll
        for (int r = 0; r < 8; ++r) {
            int ml = tile * 16 + mhalf + r;
            int bo = ml >> 9;
            int rr = ml & 511;
            int o1 = rr >> 7, o2 = (rr >> 5) & 3, o3 = (rr >> 3) & 3, o4 = rr & 7;
            float v = acc[r] + bias;
            v = v > 0.0f ? v : 0.0f;
            h1[bo * 1024 + n * 512 + o1 * 128 + o2 * 32 + o3 * 8 + o4] = v;
        }
    }
}

// ---------------------------------------------------------------------------
// Kernel B: conv2 (h1 -> 32 feats/batch) + relu + concat(label) + GEMV head.
// One wave per batch; lane = flat output position; wave32 shuffle reduction.
// ---------------------------------------------------------------------------
__global__ __launch_bounds__(128) void conv2_head(
    const float* __restrict__ h1,    // [64,2,4,4,4,8]
    const float* __restrict__ w2,    // [1,2,3,3,3,3] == [cin*81 + tap]
    const float* __restrict__ b2,    // [1]
    const float* __restrict__ label, // [64,1]
    const float* __restrict__ wout,  // [1,33]
    const float* __restrict__ bout,  // [1]
    float* __restrict__ out)         // [64,1]
{
    const int tid  = threadIdx.x;
    const int lane = tid & 31;
    const int bidx = blockIdx.x * 4 + (tid >> 5);   // batch 0..63

    // lane == flat position p in h2 [1,2,2,2,4]
    const int p   = lane;
    const int i1b = ( p >> 4      ) * 2 - 1;
    const int i2b = ((p >> 3) & 1 ) * 2 - 1;
    const int i3b = ((p >> 2) & 1 ) * 2 - 1;
    const int i4b = ( p       & 3 ) * 2 - 1;
    const float* hb = h1 + bidx * 1024;

    float acc = b2[0];
#pragma unroll
    for (int cin = 0; cin < 2; ++cin) {
        const float* hc = hb + cin * 512;
        const float* wc = w2 + cin * 81;
#pragma unroll
        for (int k1 = 0; k1 < 3; ++k1) {
            int i1 = i1b + k1; if ((unsigned)i1 >= 4u) continue;
#pragma unroll
            for (int k2 = 0; k2 < 3; ++k2) {
                int i2 = i2b + k2; if ((unsigned)i2 >= 4u) continue;
#pragma unroll
                for (int k3 = 0; k3 < 3; ++k3) {
                    int i3 = i3b + k3; if ((unsigned)i3 >= 4u) continue;
#pragma unroll
                    for (int k4 = 0; k4 < 3; ++k4) {
                        int i4 = i4b + k4; if ((unsigned)i4 >= 8u) continue;
                        acc += hc[i1 * 128 + i2 * 32 + i3 * 8 + i4] *
                               wc[k1 * 27 + k2 * 9 + k3 * 3 + k4];
                    }
                }
            }
        }
    }
    acc = acc > 0.0f ? acc : 0.0f;               // relu
    float part = acc * wout[p];                  // head partial product
    for (int off = 16; off > 0; off >>= 1)       // wave32 reduction
        part += __shfl_down(part, off, 32);
    if (lane == 0)
        out[bidx] = part + label[bidx] * wout[32] + bout[0];
}

// ---------------------------------------------------------------------------
extern "C" void kernel_launch(void* const* d_in, const int* in_sizes, int n_in,
                              void* d_out, int out_size, void* d_ws, size_t ws_size,
                              hipStream_t stream) {
    const float* x     = (const float*)d_in[0];  // [64,4,8,8,8,16]
    const float* label = (const float*)d_in[1];  // [64,1]
    const float* w1    = (const float*)d_in[2];  // [2,4,3,3,3,3]
    const float* b1    = (const float*)d_in[3];  // [2]
    const float* w2    = (const float*)d_in[4];  // [1,2,3,3,3,3]
    const float* b2    = (const float*)d_in[5];  // [1]
    const float* wout  = (const float*)d_in[6];  // [1,33]
    const float* bout  = (const float*)d_in[7];  // [1]
    float*       out   = (float*)d_out;          // [64,1]
    float*       h1    = (float*)d_ws;           // 65536 floats = 256 KB

    // conv1: 2048 M-tiles, 4 waves/block -> 512 blocks x 128 threads
    conv1_wmma_f32<<<512, 128, 0, stream>>>(x, w1, b1, h1);
    // conv2 + head: 64 batches, 1 wave each -> 16 blocks x 128 threads
    conv2_head<<<16, 128, 0, stream>>>(h1, w2, b2, label, wout, bout, out);
}